// PSigned_9113920602642
// MI455X (gfx1250) — compile-verified
//
#include <hip/hip_runtime.h>
#include <cmath>

typedef __attribute__((ext_vector_type(2))) float v2f;
typedef __attribute__((ext_vector_type(8))) float v8f;

#define NNODES 50000
#define NEDGES 800000
#define FDIM   128     // feature dim of every layer input (F == H == 128)
#define KDIM   256     // concat dim: [self || aggr]
#define LDS_STRIDE 260 // 16B-aligned row stride, bank-conflict-free A reads

// ---------------------------------------------------------------- zero aggr
__global__ void PS_zero_kernel(float4* __restrict__ p, int n4) {
    int i = blockIdx.x * blockDim.x + threadIdx.x;
    if (i < n4) p[i] = make_float4(0.f, 0.f, 0.f, 0.f);
}

// ------------------------------------------- transpose W [KDIM,nOut] -> Wt [nOut,KDIM]
// one-time per layer (<=128KB); makes B fragments contiguous b64 loads.
__global__ void PS_transpose_kernel(const float* __restrict__ W,
                                    float* __restrict__ Wt, int nOut) {
    int idx = blockIdx.x * blockDim.x + threadIdx.x;
    if (idx < KDIM * nOut) {
        int k = idx / nOut;
        int n = idx - k * nOut;          // consecutive threads: consecutive n -> coalesced read
        Wt[(size_t)n * KDIM + k] = W[idx];
    }
}

// ------------------------------------------------- scatter-add over edges
// one wave32 per edge; each lane handles 4 consecutive features.
// edge id is computed wave-uniformly so src/dst become scalar loads.
__global__ void PS_scatter_kernel(const float* __restrict__ h,
                                  const int*  __restrict__ ei,
                                  float* __restrict__ aggr, int E) {
    int edge = blockIdx.x * 8 + (threadIdx.x >> 5);   // 8 waves / 256-thr block
    int lane = threadIdx.x & 31;
    if (edge >= E) return;
    int src = ei[edge];
    int dst = ei[E + edge];
    const float4 v = *(const float4*)(h + (size_t)src * FDIM + lane * 4);
    float* o = aggr + (size_t)dst * FDIM + lane * 4;
    atomicAdd(o + 0, v.x);
    atomicAdd(o + 1, v.y);
    atomicAdd(o + 2, v.z);
    atomicAdd(o + 3, v.w);
}

// ------------------------------- fused [self||aggr] @ W + b, activation
// block = 256 thr (8 waves) -> one 16-row tile; wave w -> 16-col tile w.
// K loop: V_WMMA_F32_16X16X4_F32, A from LDS (ds b64), B from transposed W
// (global b64, fixed base + immediate offsets). Full fp32 accumulate.
template <int NOUT, int ACT>
__global__ void PS_gemm_kernel(const float* __restrict__ hSelf,
                               const float* __restrict__ hAggr,
                               const float* __restrict__ Wt,   // [NOUT, KDIM]
                               const float* __restrict__ bias,
                               float* __restrict__ out) {
    __shared__ float xs[16 * LDS_STRIDE];
    const int tid     = threadIdx.x;
    const int rowBase = blockIdx.x * 16;

    // ---- stage 16x256 concat tile into LDS (each thread: 16 floats) ----
    {
        int r  = tid >> 4;
        int c0 = (tid & 15) * 16;
        const float* srcRow = (c0 < FDIM)
            ? (hSelf + (size_t)(rowBase + r) * FDIM + c0)
            : (hAggr + (size_t)(rowBase + r) * FDIM + (c0 - FDIM));
        float* dst = xs + r * LDS_STRIDE + c0;
        #pragma unroll
        for (int j = 0; j < 4; ++j) {
            float4 v = *(const float4*)(srcRow + j * 4);
            *(float4*)(dst + j * 4) = v;
        }
    }
    __syncthreads();

    const int wave = tid >> 5;
    const int lane = tid & 31;
    constexpr int colTiles = NOUT >> 4;
    if (wave >= colTiles) return;

    const int n0 = wave * 16;
    const int m  = lane & 15;   // matrix row (A) / column (B,C,D) owned by lane
    const int hi = lane >> 4;   // upper half-wave -> K+{2,3} (A,B) / M+8 (C,D)

    v8f acc = {};
    const float* xrow = xs + m * LDS_STRIDE + 2 * hi;          // A: row m, K off 2*hi
    const float* wrow = Wt + (size_t)(n0 + m) * KDIM + 2 * hi; // B: col n0+m, contiguous K

    #pragma unroll 8
    for (int k = 0; k < KDIM; k += 4) {
        v2f a = *(const v2f*)(xrow + k);   // {A[m][k+2hi], A[m][k+2hi+1]}
        v2f b = *(const v2f*)(wrow + k);   // {B[k+2hi][n0+m], B[k+2hi+1][n0+m]}
        acc = __builtin_amdgcn_wmma_f32_16x16x4_f32(
            /*neg_a=*/false, a, /*neg_b=*/false, b,
            /*c_mod=*/(short)0, acc, /*reuse_a=*/false, /*reuse_b=*/false);
    }

    const float bcol = bias[n0 + m];
    #pragma unroll
    for (int r = 0; r < 8; ++r) {
        int row = rowBase + r + 8 * hi;   // C/D layout: vgpr r -> M=r (lo) / r+8 (hi)
        float v = acc[r] + bcol;
        v = ACT ? tanhf(v) : fmaxf(v, 0.f);
        out[(size_t)row * NOUT + (n0 + m)] = v;
    }
}

// ---------------------------------------------------------------- driver
extern "C" void kernel_launch(void* const* d_in, const int* in_sizes, int n_in,
                              void* d_out, int out_size, void* d_ws, size_t ws_size,
                              hipStream_t stream) {
    const float* x  = (const float*)d_in[0];
    const int*   ei = (const int*)  d_in[1];
    const float* W0 = (const float*)d_in[2];
    const float* b0 = (const float*)d_in[3];
    const float* W1 = (const float*)d_in[4];
    const float* b1 = (const float*)d_in[5];
    const float* W2 = (const float*)d_in[6];
    const float* b2 = (const float*)d_in[7];
    const float* W3 = (const float*)d_in[8];
    const float* b3 = (const float*)d_in[9];
    float* out = (float*)d_out;

    const size_t nodeElems = (size_t)NNODES * FDIM; // 6.4M floats
    float* aggr = (float*)d_ws;
    float* hA   = aggr + nodeElems;
    float* hB   = hA   + nodeElems;
    float* Wt   = hB   + nodeElems;                 // up to 256*128 floats

    const int zeroN4     = (int)(nodeElems / 4);
    const int zeroBlocks = (zeroN4 + 255) / 256;
    const int scatBlocks = (NEDGES + 7) / 8;        // 1 wave per edge, 8 waves/block
    const int gemmBlocks = NNODES / 16;             // 50000 % 16 == 0

    auto pre = [&](const float* hin, const float* Wl, int nOut) {
        PS_zero_kernel<<<zeroBlocks, 256, 0, stream>>>((float4*)aggr, zeroN4);
        PS_transpose_kernel<<<(KDIM * nOut + 255) / 256, 256, 0, stream>>>(Wl, Wt, nOut);
        PS_scatter_kernel<<<scatBlocks, 256, 0, stream>>>(hin, ei, aggr, NEDGES);
    };

    // conv_input + relu
    pre(x, W0, 128);
    PS_gemm_kernel<128, 0><<<gemmBlocks, 256, 0, stream>>>(x,  aggr, Wt, b0, hA);
    // hidden layer 1 + relu
    pre(hA, W1, 128);
    PS_gemm_kernel<128, 0><<<gemmBlocks, 256, 0, stream>>>(hA, aggr, Wt, b1, hB);
    // hidden layer 2 + relu
    pre(hB, W2, 128);
    PS_gemm_kernel<128, 0><<<gemmBlocks, 256, 0, stream>>>(hB, aggr, Wt, b2, hA);
    // conv_output + tanh
    pre(hA, W3, 16);
    PS_gemm_kernel<16, 1><<<gemmBlocks, 256, 0, stream>>>(hA, aggr, Wt, b3, out);
}